// SpikeFP32GELUExact_29875792511549
// MI455X (gfx1250) — compile-verified
//
#include <hip/hip_runtime.h>
#include <cmath>

#define WAVES_PER_BLOCK 8
#define THREADS_PER_BLOCK (WAVES_PER_BLOCK * 32)

// Types the async-LDS builtins expect (from clang's diagnostic):
// int vector(4) pointers in the global (1) / LDS (3) address spaces.
typedef int v4i __attribute__((vector_size(16)));
typedef __attribute__((address_space(1))) v4i glb_v4i;
typedef __attribute__((address_space(3))) v4i lds_v4i;

// ---- CDNA5 async global<->LDS helpers (ASYNCcnt path) ----------------------

__device__ __forceinline__ void async_copy_g2l_b128(const void* gsrc, void* ldst) {
#if __has_builtin(__builtin_amdgcn_global_load_async_to_lds_b128)
    __builtin_amdgcn_global_load_async_to_lds_b128(
        (glb_v4i*)gsrc, (lds_v4i*)ldst, 0, 0);
#else
    unsigned loff = (unsigned)(__SIZE_TYPE__)(lds_v4i*)ldst;
    asm volatile("global_load_async_to_lds_b128 %0, %1, off"
                 :: "v"(loff), "v"(gsrc) : "memory");
#endif
}

__device__ __forceinline__ void async_copy_l2g_b128(void* gdst, const void* lsrc) {
#if __has_builtin(__builtin_amdgcn_global_store_async_from_lds_b128)
    __builtin_amdgcn_global_store_async_from_lds_b128(
        (glb_v4i*)gdst, (lds_v4i*)lsrc, 0, 0);
#else
    unsigned loff = (unsigned)(__SIZE_TYPE__)(lds_v4i*)lsrc;
    asm volatile("global_store_async_from_lds_b128 %0, %1, off"
                 :: "v"(gdst), "v"(loff) : "memory");
#endif
}

__device__ __forceinline__ void wait_async0() {
    asm volatile("" ::: "memory");
#if __has_builtin(__builtin_amdgcn_s_wait_asynccnt)
    __builtin_amdgcn_s_wait_asynccnt(0);
#else
    asm volatile("s_wait_asynccnt 0x0" ::: "memory");
#endif
    asm volatile("" ::: "memory");
}

__device__ __forceinline__ void wait_ds0() {
    asm volatile("" ::: "memory");
#if __has_builtin(__builtin_amdgcn_s_wait_dscnt)
    __builtin_amdgcn_s_wait_dscnt(0);
#else
    asm volatile("s_wait_dscnt 0x0" ::: "memory");
#endif
    asm volatile("" ::: "memory");
}

// ---- exact fp64 tanh-GELU matching the reference op order ------------------

__device__ __forceinline__ unsigned gelu_bits(unsigned uin) {
#pragma clang fp contract(off)
    const double v   = (double)__uint_as_float(uin);
    const double xsq = v * v;
    const double xcb = xsq * v;
    const double inn = v + 0.044715 * xcb;
    const double z   = 0.7978845608028654 * inn;
    const double e2z = exp(2.0 * z);
    const double th  = (e2z - 1.0) / (e2z + 1.0);
    const double r   = 0.5 * (v * (1.0 + th));
    return __float_as_uint((float)r);  // RNE down-cast
}

// ---- kernel ----------------------------------------------------------------

__global__ __launch_bounds__(THREADS_PER_BLOCK)
void SpikeFP32GELUExact_29875792511549_kernel(const float* __restrict__ x,
                                              float* __restrict__ out,
                                              long long nvals) {
    __shared__ float ldsIn [WAVES_PER_BLOCK][1024];   // 4 KB per wave
    __shared__ float ldsOut[WAVES_PER_BLOCK][1024];   // 4 KB per wave

    const int lane = threadIdx.x & 31;
    const int wv   = threadIdx.x >> 5;

    float* bin  = ldsIn [wv];
    float* bout = ldsOut[wv];

    const long long nchunks = (nvals + 31) >> 5;
    const long long gwave   = (long long)blockIdx.x * WAVES_PER_BLOCK + wv;
    const long long gstride = (long long)gridDim.x * WAVES_PER_BLOCK;

    for (long long c = gwave; c < nchunks; c += gstride) {
        const long long vbase = c << 5;                  // first value of chunk

        if (vbase + 32 <= nvals) {
            // ---------------- fast path: full 32-value chunk ----------------
            const float* gsrc = x   + (vbase << 5);      // 1024 floats (4 KB)
            float*       gdst = out + (vbase << 5);

            // Stage input tile into LDS: 8 x (32 lanes x 16B) async copies.
            #pragma unroll
            for (int r = 0; r < 8; ++r) {
                const int q = r * 32 + lane;             // 16-byte chunk index
                async_copy_g2l_b128(gsrc + q * 4, bin + q * 4);
            }
            // Drains these loads AND the previous iteration's stores, so both
            // bin (now refilled) and bout (about to be rewritten) are safe.
            wait_async0();

            // Pack: lane l owns value (vbase+l). Swizzled order j=(i+l)&31
            // -> LDS bank (33*l+i) mod 64: conflict-free across the wave.
            unsigned u = 0u;
            #pragma unroll
            for (int i = 0; i < 32; ++i) {
                const int j = (i + lane) & 31;
                const unsigned bit = (unsigned)bin[lane * 32 + j]; // 0.0/1.0 -> 0/1
                u |= bit << (31 - j);                              // element j = bit (31-j)
            }

            const unsigned uo = gelu_bits(u);

            // Unpack result bits into the LDS out-tile (same swizzle).
            #pragma unroll
            for (int i = 0; i < 32; ++i) {
                const int j = (i + lane) & 31;
                bout[lane * 32 + j] = (float)((uo >> (31 - j)) & 1u);
            }

            // LDS writes must land before the async engine reads them.
            wait_ds0();

            // Stream the out-tile back with async LDS->global b128 copies.
            #pragma unroll
            for (int r = 0; r < 8; ++r) {
                const int q = r * 32 + lane;
                async_copy_l2g_b128(gdst + q * 4, bout + q * 4);
            }
            // No wait here: next iteration's wait_async0 (or the implicit
            // S_WAIT_IDLE at S_ENDPGM) covers these stores.
        } else {
            // ---------------- tail path: per-lane, bounds-checked -----------
            const long long v = vbase + lane;
            if (v < nvals) {
                const float* p = x + (v << 5);
                unsigned u = 0u;
                for (int i = 0; i < 32; ++i)
                    u |= ((unsigned)p[i]) << (31 - i);
                const unsigned uo = gelu_bits(u);
                float* q = out + (v << 5);
                for (int i = 0; i < 32; ++i)
                    q[i] = (float)((uo >> (31 - i)) & 1u);
            }
        }
    }
}

// ---- launch ----------------------------------------------------------------

extern "C" void kernel_launch(void* const* d_in, const int* in_sizes, int n_in,
                              void* d_out, int out_size, void* d_ws, size_t ws_size,
                              hipStream_t stream) {
    (void)n_in; (void)out_size; (void)d_ws; (void)ws_size;

    const float* x  = (const float*)d_in[0];
    float*       o  = (float*)d_out;

    const long long nbits   = (long long)in_sizes[0];
    const long long nvals   = nbits / 32;                 // 4,194,304 for B,S,D=2,2048,1024
    const long long nchunks = (nvals + 31) / 32;

    long long blocks = (nchunks + WAVES_PER_BLOCK - 1) / WAVES_PER_BLOCK;
    if (blocks < 1) blocks = 1;
    if (blocks > 262144) blocks = 262144;                 // grid-stride covers the rest

    SpikeFP32GELUExact_29875792511549_kernel<<<(int)blocks, THREADS_PER_BLOCK, 0, stream>>>(
        x, o, nvals);
}